// IWT_12463995093517
// MI455X (gfx1250) — compile-verified
//
#include <hip/hip_runtime.h>

// Inverse wavelet transform for x:[32,64,128,128] f32 -> out:[8,64,256,256] f32.
// Pure bandwidth problem (~268 MB traffic, ~11.5 us floor at 23.3 TB/s), so the
// 4x4 Haar butterfly is folded into V_WMMA_F32_16X16X4_F32 (A = +/-0.5 coeff
// matrix, B = 4 subband values per column) purely because the D-matrix layout
// hands back {e00,e01} to lanes 0-15 and {e10,e11} to lanes 16-31, giving
// contiguous 128B b64 stores per half-wave with no cross-lane shuffles.

typedef __attribute__((ext_vector_type(2))) float v2f;
typedef __attribute__((ext_vector_type(8))) float v8f;

#define SUB_OFF (8 * 64 * 128 * 128)   // elements per subband block (x1..x4)

__global__ __launch_bounds__(256) void iwt_wmma_kernel(const float* __restrict__ x,
                                                       float* __restrict__ out)
{
    const int tid  = threadIdx.x;
    const int lane = tid & 31;
    const int wave = tid >> 5;
    const int row  = blockIdx.x * 8 + wave;     // 0 .. 65535 == (b*64+c)*128 + i

    const int half = lane >> 4;                 // 0: subbands 1,2 / even out-row
                                                // 1: subbands 3,4 / odd  out-row
    const int n    = lane & 15;                 // column within 16-wide chunk

    const int i  = row & 127;                   // H = 128
    const int bc = row >> 7;                    // b*64 + c

    const int base_in  = (bc * 128 + i) * 128;                   // x1-plane row start
    const int base_out = (bc * 256 + 2 * i + half) * 256;        // output row start

    // ---- A matrix: A[m][k] = 0.5 * sign(m&3, k) ------------------------------
    // row r -> (p,q) = (r>>1, r&1); out pixel (2i+p, 2j+q)
    //   k=0 (x1): +        k=1 (x2): q ? + : -
    //   k=2 (x3): p ? + : -  k=3 (x4): (p==q) ? + : -
    // A layout (16x4 f32): lanes 0-15 = {K0,K1} at M=lane; lanes 16-31 = {K2,K3}.
    const int r = n & 3;
    const int q = r & 1;
    const int p = (r >> 1) & 1;
    const float s1 = q ? 0.5f : -0.5f;
    const float s2 = p ? 0.5f : -0.5f;
    const float s3 = (p == q) ? 0.5f : -0.5f;
    v2f A;
    A.x = half ? s2 : 0.5f;    // K = 2*half
    A.y = half ? s3 : s1;      // K = 2*half + 1

    // Per-lane input base: lanes 0-15 read x1/x2 planes, lanes 16-31 read x3/x4.
    const float* p0 = x + base_in + (2 * half) * SUB_OFF + n;   // K = 2*half
    const float* p1 = p0 + SUB_OFF;                             // K = 2*half + 1
    float* po = out + base_out;

    #pragma unroll
    for (int j0 = 0; j0 < 128; j0 += 16) {
        // ---- B matrix (4x16 f32): lanes 0-15 = {K0,K1} at N=lane; 16-31 = {K2,K3}
        v2f B;
        B.x = p0[j0];          // 64B contiguous per half-wave per subband plane
        B.y = p1[j0];

        v8f c = {};
        // D[m][nn] = sum_k A[m][k] * B[k][nn]  ->  VGPR v = e_{v&3}[nn],
        // replicated into both half-waves of the wave.
        c = __builtin_amdgcn_wmma_f32_16x16x4_f32(
                /*neg_a=*/false, A, /*neg_b=*/false, B,
                /*c_mod=*/(short)0, c, /*reuse_a=*/false, /*reuse_b=*/false);

        // lanes 0-15: {e00,e01} -> row 2i ; lanes 16-31: {e10,e11} -> row 2i+1
        v2f o;
        o.x = half ? c[2] : c[0];
        o.y = half ? c[3] : c[1];
        *(v2f*)(po + 2 * (j0 + n)) = o;   // global_store_b64, 128B per half-wave
    }
}

extern "C" void kernel_launch(void* const* d_in, const int* in_sizes, int n_in,
                              void* d_out, int out_size, void* d_ws, size_t ws_size,
                              hipStream_t stream) {
    const float* x = (const float*)d_in[0];
    float* out = (float*)d_out;
    // rows = B*C*H = 8*64*128 = 65536 ; 8 waves (rows) per 256-thread block
    const int blocks = 65536 / 8;
    iwt_wmma_kernel<<<blocks, 256, 0, stream>>>(x, out);
}